// RNNModel_4045859193155
// MI455X (gfx1250) — compile-verified
//
#include <hip/hip_runtime.h>
#include <hip/hip_bf16.h>

// ---------------------------------------------------------------------------
// Fused GRU-GRU-GLU-Dense inference for MI455X (gfx1250, wave32, WMMA).
// One wave (32 lanes) per block owns a 16-row batch tile and runs the whole
// T=1024 recurrence with state + ALL recurrent weights in LDS, using
// V_WMMA_F32_16X16X4_F32 for every matmul tile. LDS strides are padded so
// every A/B fragment load is 64-bank conflict-free.
// ---------------------------------------------------------------------------

typedef float v2f __attribute__((ext_vector_type(2)));
typedef float v8f __attribute__((ext_vector_type(8)));

#define VOCAB  50257
#define DIM    50
#define DPAD   52          // 50 padded to multiple of 4 (also bank-friendly)
#define U1     32
#define G1     96          // 3*U1
#define U2     64
#define G2     192         // 3*U2
#define GLU_U  128
#define BATCH  1024
#define TSEQ   1024

// Padded LDS strides (floats). Chosen so that:
//  - A-frag loads:  (lane&15)*LDA covers 16 distinct banks, all = 0 (mod 4)
//  - B-frag loads:  2*LDB = 16 (mod 64)  -> lane halves hit disjoint banks
#define SX     52          // x tile stride       (52*l mod 64: 16 distinct)
#define SH1    36          // h1 tile stride      (36*l mod 64: 16 distinct)
#define SH2    68          // h2 tile stride      (68*l mod 64: 16 distinct)
#define SW1    104         // kx1/kh1 row stride  (208 mod 64 = 16)
#define SW2    200         // kx2/kh2 row stride  (400 mod 64 = 16)

__device__ __forceinline__ float sigf(float x) {
    return 1.0f / (1.0f + __expf(-x));
}
__device__ __forceinline__ float tanhfast(float x) {
    float e = __expf(-2.0f * x);
    return (1.0f - e) / (1.0f + e);
}

// One 16x16 output tile: acc += A[16 x 4K] * B[4K x 16-col-slice n].
// Fragment layout per CDNA5 ISA 7.12.2 (16x16x4 f32):
//   A: lane lm holds row M=lm, K pair (4k+2*half, 4k+2*half+1) -> one b64 load
//   B: .x = row (4k+2*half), .y = row (+1), col = 16n+lm
template <int KSTEPS>
__device__ __forceinline__ v8f mm_tile(const float* A, int lda,
                                       const float* B, int ldb,
                                       int n, v8f acc, int lm, int half) {
#pragma unroll
    for (int k = 0; k < KSTEPS; ++k) {
        v2f a = *(const v2f*)(A + lm * lda + 4 * k + 2 * half);
        int r = 4 * k + 2 * half;
        v2f b;
        b.x = B[r * ldb + 16 * n + lm];
        b.y = B[(r + 1) * ldb + 16 * n + lm];
        acc = __builtin_amdgcn_wmma_f32_16x16x4_f32(
            /*neg_a=*/false, a, /*neg_b=*/false, b,
            /*c_mod=*/(short)0, acc, /*reuse_a=*/false, /*reuse_b=*/false);
    }
    return acc;
}

__global__ __launch_bounds__(32, 1) void rnn_fused_kernel(
    const int*   __restrict__ tokens,  // [B,T]
    const float* __restrict__ emb,     // [V,50]
    const float* __restrict__ kx1,     // [50,96]
    const float* __restrict__ kh1,     // [32,96]
    const float* __restrict__ b1,      // [2,96]
    const float* __restrict__ kx2,     // [32,192]
    const float* __restrict__ kh2,     // [64,192]
    const float* __restrict__ b2,      // [2,192]
    const float* __restrict__ wg,      // [64,256]
    const float* __restrict__ bg,      // [256]
    const float* __restrict__ wd,      // [128,1]
    const float* __restrict__ bd,      // [1]
    float*       __restrict__ out)     // [B,1]
{
    // ~121 KB total static LDS: one workgroup per WGP (320 KB available).
    __shared__ __align__(16) float lds_kx1[DPAD * SW1];  // 21.1 KB (rows 50,51 zero)
    __shared__ __align__(16) float lds_kh1[U1 * SW1];    // 13.0 KB
    __shared__ __align__(16) float lds_kx2[U1 * SW2];    // 25.0 KB
    __shared__ __align__(16) float lds_kh2[U2 * SW2];    // 50.0 KB
    __shared__ __align__(16) float lds_b1[2 * G1];
    __shared__ __align__(16) float lds_b2[2 * G2];
    __shared__ __align__(16) float lds_x[16 * SX];       // x_t tile, cols 50,51 zero
    __shared__ __align__(16) float lds_h1[16 * SH1];     // GRU1 state (C-layout rows)
    __shared__ __align__(16) float lds_h2[16 * SH2];     // GRU2 state
    __shared__ __align__(16) float lds_out[16];

    const int lane = threadIdx.x;      // 0..31, single wave32
    const int lm   = lane & 15;
    const int half = lane >> 4;
    const int r0   = blockIdx.x * 16;  // batch rows [r0, r0+16)

    // ---- one-time weight staging + state init (cost amortized over T=1024) --
    for (int i = lane; i < DPAD * G1; i += 32) {
        int r = i / G1, c = i % G1;
        lds_kx1[r * SW1 + c] = (r < DIM) ? kx1[i] : 0.0f;
    }
    for (int i = lane; i < U1 * G1; i += 32) {
        int r = i / G1, c = i % G1;
        lds_kh1[r * SW1 + c] = kh1[i];
    }
    for (int i = lane; i < U1 * G2; i += 32) {
        int r = i / G2, c = i % G2;
        lds_kx2[r * SW2 + c] = kx2[i];
    }
    for (int i = lane; i < U2 * G2; i += 32) {
        int r = i / G2, c = i % G2;
        lds_kh2[r * SW2 + c] = kh2[i];
    }
    for (int i = lane; i < 2 * G1;  i += 32) lds_b1[i] = b1[i];
    for (int i = lane; i < 2 * G2;  i += 32) lds_b2[i] = b2[i];
    for (int i = lane; i < 16 * SH1; i += 32) lds_h1[i] = 0.0f;
    for (int i = lane; i < 16 * SH2; i += 32) lds_h2[i] = 0.0f;
    if (lane < 16) lds_out[lane] = 0.0f;
    __syncthreads();

    const v8f Z8 = {0.f, 0.f, 0.f, 0.f, 0.f, 0.f, 0.f, 0.f};

    // ---- serial scan over time ----------------------------------------------
    for (int t = 0; t < TSEQ; ++t) {
        // Gather embedding tile x_t [16 x 50] -> LDS (padded to 52 cols).
        int tok = tokens[(size_t)(r0 + lm) * TSEQ + t];
        const float* erow = emb + (size_t)tok * DIM;
#pragma unroll
        for (int i = 0; i < 13; ++i) {
            int d0 = half * 26 + 2 * i;       // lo lanes dims 0..25, hi 26..51
            v2f xv = {0.f, 0.f};
            if (d0 < DIM) xv = *(const v2f*)(erow + d0);
            *(v2f*)(&lds_x[lm * SX + d0]) = xv;
        }
        // Prefetch next step's embedding row into cache (global_prefetch_b8).
        if (t + 1 < TSEQ) {
            int tok2 = tokens[(size_t)(r0 + lm) * TSEQ + t + 1];
            __builtin_prefetch(emb + (size_t)tok2 * DIM, 0, 1);
        }
        __syncthreads();

        // ---------------- GRU1: h1 = gru(x_t, h1) ; U1=32 -> 2 N-tiles -------
        v8f h1n[2];
#pragma unroll
        for (int j = 0; j < 2; ++j) {
            v8f xz = mm_tile<13>(lds_x,  SX,  lds_kx1, SW1, j,     Z8, lm, half);
            v8f hz = mm_tile<8>(lds_h1,  SH1, lds_kh1, SW1, j,     Z8, lm, half);
            v8f xr = mm_tile<13>(lds_x,  SX,  lds_kx1, SW1, 2 + j, Z8, lm, half);
            v8f hr = mm_tile<8>(lds_h1,  SH1, lds_kh1, SW1, 2 + j, Z8, lm, half);
            v8f xh = mm_tile<13>(lds_x,  SX,  lds_kx1, SW1, 4 + j, Z8, lm, half);
            v8f hm = mm_tile<8>(lds_h1,  SH1, lds_kh1, SW1, 4 + j, Z8, lm, half);
            int c = 16 * j + lm;
            float bz  = lds_b1[c],            bzh = lds_b1[G1 + c];
            float br  = lds_b1[U1 + c],       brh = lds_b1[G1 + U1 + c];
            float bhx = lds_b1[2 * U1 + c],   bhh = lds_b1[G1 + 2 * U1 + c];
#pragma unroll
            for (int m = 0; m < 8; ++m) {
                float hold = lds_h1[(m + 8 * half) * SH1 + c];
                float z  = sigf(xz[m] + bz + hz[m] + bzh);
                float r  = sigf(xr[m] + br + hr[m] + brh);
                float hc = tanhfast(xh[m] + bhx + r * (hm[m] + bhh));
                h1n[j][m] = z * hold + (1.0f - z) * hc;
            }
        }
#pragma unroll
        for (int j = 0; j < 2; ++j)
#pragma unroll
            for (int m = 0; m < 8; ++m)
                lds_h1[(m + 8 * half) * SH1 + 16 * j + lm] = h1n[j][m];

        // ---------------- GRU2: h2 = gru(h1, h2) ; U2=64 -> 4 N-tiles --------
        v8f h2n[4];
#pragma unroll
        for (int j = 0; j < 4; ++j) {
            v8f xz = mm_tile<8>(lds_h1,  SH1, lds_kx2, SW2, j,     Z8, lm, half);
            v8f hz = mm_tile<16>(lds_h2, SH2, lds_kh2, SW2, j,     Z8, lm, half);
            v8f xr = mm_tile<8>(lds_h1,  SH1, lds_kx2, SW2, 4 + j, Z8, lm, half);
            v8f hr = mm_tile<16>(lds_h2, SH2, lds_kh2, SW2, 4 + j, Z8, lm, half);
            v8f xh = mm_tile<8>(lds_h1,  SH1, lds_kx2, SW2, 8 + j, Z8, lm, half);
            v8f hm = mm_tile<16>(lds_h2, SH2, lds_kh2, SW2, 8 + j, Z8, lm, half);
            int c = 16 * j + lm;
            float bz  = lds_b2[c],            bzh = lds_b2[G2 + c];
            float br  = lds_b2[U2 + c],       brh = lds_b2[G2 + U2 + c];
            float bhx = lds_b2[2 * U2 + c],   bhh = lds_b2[G2 + 2 * U2 + c];
#pragma unroll
            for (int m = 0; m < 8; ++m) {
                float hold = lds_h2[(m + 8 * half) * SH2 + c];
                float z  = sigf(xz[m] + bz + hz[m] + bzh);
                float r  = sigf(xr[m] + br + hr[m] + brh);
                float hc = tanhfast(xh[m] + bhx + r * (hm[m] + bhh));
                h2n[j][m] = z * hold + (1.0f - z) * hc;
            }
        }
#pragma unroll
        for (int j = 0; j < 4; ++j)
#pragma unroll
            for (int m = 0; m < 8; ++m)
                lds_h2[(m + 8 * half) * SH2 + 16 * j + lm] = h2n[j][m];

        __syncthreads();
    }

    // ---- GLU + dense head: out = sigmoid((a[:,:128] * sig(a[:,128:])) @ wd + bd)
    // One-shot epilogue: wg streams from L2 (64 KB, shared by all blocks).
#pragma unroll
    for (int j = 0; j < 8; ++j) {
        v8f a1 = mm_tile<16>(lds_h2, SH2, wg, 2 * GLU_U, j,     Z8, lm, half);
        v8f a2 = mm_tile<16>(lds_h2, SH2, wg, 2 * GLU_U, 8 + j, Z8, lm, half);
        float bga = bg[16 * j + lm];
        float bgb = bg[GLU_U + 16 * j + lm];
        float w   = wd[16 * j + lm];
#pragma unroll
        for (int m = 0; m < 8; ++m) {
            float g = (a1[m] + bga) * sigf(a2[m] + bgb);
            atomicAdd(&lds_out[m + 8 * half], g * w);
        }
    }
    __syncthreads();
    if (lane < 16) out[r0 + lane] = sigf(lds_out[lane] + bd[0]);
}

extern "C" void kernel_launch(void* const* d_in, const int* in_sizes, int n_in,
                              void* d_out, int out_size, void* d_ws, size_t ws_size,
                              hipStream_t stream) {
    (void)in_sizes; (void)n_in; (void)d_ws; (void)ws_size; (void)out_size;
    const int*   tokens = (const int*)  d_in[0];
    const float* emb    = (const float*)d_in[1];
    const float* kx1    = (const float*)d_in[2];
    const float* kh1    = (const float*)d_in[3];
    const float* b1     = (const float*)d_in[4];
    const float* kx2    = (const float*)d_in[5];
    const float* kh2    = (const float*)d_in[6];
    const float* b2     = (const float*)d_in[7];
    const float* wg     = (const float*)d_in[8];
    const float* bg     = (const float*)d_in[9];
    const float* wd     = (const float*)d_in[10];
    const float* bd     = (const float*)d_in[11];
    float* out = (float*)d_out;

    dim3 grid(BATCH / 16);   // 64 blocks, one 16-row batch tile each
    dim3 block(32);          // one wave32 per block
    rnn_fused_kernel<<<grid, block, 0, stream>>>(
        tokens, emb, kx1, kh1, b1, kx2, kh2, b2, wg, bg, wd, bd, out);
}